// MomentumContrast_36163624632945
// MI455X (gfx1250) — compile-verified
//
#include <hip/hip_runtime.h>
#include <math.h>

typedef __attribute__((ext_vector_type(2))) float v2f;
typedef __attribute__((ext_vector_type(8))) float v8f;

#define TEMP_INV 14.285714285714286f   /* 1/0.07 */
#define B_    1024
#define DIN_  2048
#define H_    1024
#define M_    256
#define Q_    65536
#define NNEG_ 16384
#define NT_   256                      /* NNEG / 64 column tiles */

// ---------------------------------------------------------------------------
// C[M,N] = A[M,K] @ B[K,N], all row-major fp32, via V_WMMA_F32_16X16X4_F32.
// 1 wave per block; each wave owns a 16x64 output tile (4 accumulators so the
// A fragment is reused 4x per K-step).
// A fragment layout (16x4 f32): lane = {hi=lane>>4, lo=lane&15};
//   a.x = A[row0+lo][k0+2*hi], a.y = A[row0+lo][k0+2*hi+1]
// B fragment (4x16): b.x = B[k0+2*hi][n], b.y = B[k0+2*hi+1][n]
// D layout: acc[r] on lane -> C[row0 + r + 8*hi][col0 + lo]
// ---------------------------------------------------------------------------
__global__ __launch_bounds__(32)
void gemm_f32_wmma(const float* __restrict__ A, const float* __restrict__ Bm,
                   float* __restrict__ C, int Kd, int Nd) {
  const int lane = threadIdx.x & 31;
  const int hi = lane >> 4, lo = lane & 15;
  const int row0 = blockIdx.y << 4;
  const int col0 = blockIdx.x << 6;

  v8f acc[4];
#pragma unroll
  for (int j = 0; j < 4; ++j) acc[j] = (v8f){0.f,0.f,0.f,0.f,0.f,0.f,0.f,0.f};

  const float* arow = A + (size_t)(row0 + lo) * Kd + 2 * hi;

  for (int k0 = 0; k0 < Kd; k0 += 4) {
    v2f a = *(const v2f*)(arow + k0);                       // 8B aligned
    const float* b0 = Bm + (size_t)(k0 + 2 * hi) * Nd + col0 + lo;
    const float* b1 = b0 + Nd;
#pragma unroll
    for (int j = 0; j < 4; ++j) {
      v2f b;
      b.x = b0[j << 4];
      b.y = b1[j << 4];
      acc[j] = __builtin_amdgcn_wmma_f32_16x16x4_f32(
          false, a, false, b, (short)0, acc[j], false, false);
    }
  }

#pragma unroll
  for (int j = 0; j < 4; ++j) {
    const int cc = col0 + (j << 4) + lo;
#pragma unroll
    for (int r = 0; r < 8; ++r) {
      const int rr = row0 + r + 8 * hi;
      C[(size_t)rr * Nd + cc] = acc[j][r];
    }
  }
}

// ---------------------------------------------------------------------------
// In-place row L2-normalize of P[rows, 256]; one block (256 thr) per row.
// ---------------------------------------------------------------------------
__global__ __launch_bounds__(256)
void rownorm_inplace(float* __restrict__ P) {
  __shared__ float sd[256];
  const int i = blockIdx.x, t = threadIdx.x;
  float v = P[(size_t)i * M_ + t];
  sd[t] = v * v;
  __syncthreads();
#pragma unroll
  for (int s = 128; s > 0; s >>= 1) {
    if (t < s) sd[t] += sd[t + s];
    __syncthreads();
  }
  const float scale = 1.f / fmaxf(sqrtf(sd[0]), 1e-12f);
  P[(size_t)i * M_ + t] = v * scale;
}

// ---------------------------------------------------------------------------
// neg[i,:] = l2norm(queue[sample_idx[i],:]); one block per negative row.
// ---------------------------------------------------------------------------
__global__ __launch_bounds__(256)
void gather_norm(const float* __restrict__ queue, const int* __restrict__ idx,
                 float* __restrict__ neg) {
  __shared__ float sd[256];
  const int i = blockIdx.x, t = threadIdx.x;
  const int src = idx[i];
  float v = queue[(size_t)src * M_ + t];
  sd[t] = v * v;
  __syncthreads();
#pragma unroll
  for (int s = 128; s > 0; s >>= 1) {
    if (t < s) sd[t] += sd[t + s];
    __syncthreads();
  }
  const float scale = 1.f / fmaxf(sqrtf(sd[0]), 1e-12f);
  neg[(size_t)i * M_ + t] = v * scale;
}

// ---------------------------------------------------------------------------
// Fused logits tile + online-softmax partials.
// Each block (1 wave) computes a 16x64 tile of (q @ neg^T)/T with WMMA over
// K = 256, then reduces each row of the tile to (rowmax, sum exp(z - rowmax))
// and writes partials to pm/ps at [row, tile_x]. Logits never hit HBM.
// neg is row-major [NNEG, 256], so the B fragment b = negT[k][n] = neg[n][k]
// is a contiguous float2 load from neg's row n.
// ---------------------------------------------------------------------------
__global__ __launch_bounds__(32)
void logits_partial(const float* __restrict__ q, const float* __restrict__ neg,
                    float* __restrict__ pm, float* __restrict__ ps) {
  __shared__ float tile[16][64];
  const int lane = threadIdx.x & 31;
  const int hi = lane >> 4, lo = lane & 15;
  const int row0 = blockIdx.y << 4;
  const int col0 = blockIdx.x << 6;

  v8f acc[4];
#pragma unroll
  for (int j = 0; j < 4; ++j) acc[j] = (v8f){0.f,0.f,0.f,0.f,0.f,0.f,0.f,0.f};

  const float* arow = q + (size_t)(row0 + lo) * M_ + 2 * hi;

  for (int k0 = 0; k0 < M_; k0 += 4) {
    v2f a = *(const v2f*)(arow + k0);
#pragma unroll
    for (int j = 0; j < 4; ++j) {
      const int n = col0 + (j << 4) + lo;
      v2f b = *(const v2f*)(neg + (size_t)n * M_ + k0 + 2 * hi); // contiguous
      acc[j] = __builtin_amdgcn_wmma_f32_16x16x4_f32(
          false, a, false, b, (short)0, acc[j], false, false);
    }
  }

#pragma unroll
  for (int j = 0; j < 4; ++j)
#pragma unroll
    for (int r = 0; r < 8; ++r)
      tile[r + 8 * hi][(j << 4) + lo] = acc[j][r];
  __syncthreads();

  if (lane < 16) {
    const int row = lane;
    float m = -3.402823466e+38f;
#pragma unroll 8
    for (int c = 0; c < 64; ++c) m = fmaxf(m, tile[row][c] * TEMP_INV);
    float s = 0.f;
#pragma unroll 8
    for (int c = 0; c < 64; ++c) s += __expf(tile[row][c] * TEMP_INV - m);
    const size_t o = (size_t)(row0 + row) * NT_ + blockIdx.x;
    pm[o] = m;
    ps[o] = s;
  }
}

// ---------------------------------------------------------------------------
// Per-row loss: pos = dot(q_i,k_i)/T; combine NT_=256 softmax partials plus
// the positive logit; loss_i = logsumexp - pos_z. Fixed-order tree reductions.
// ---------------------------------------------------------------------------
__global__ __launch_bounds__(256)
void row_loss(const float* __restrict__ q, const float* __restrict__ k,
              const float* __restrict__ pm, const float* __restrict__ ps,
              float* __restrict__ loss) {
  __shared__ float sd[256];
  const int i = blockIdx.x, t = threadIdx.x;

  // positive logit
  sd[t] = q[(size_t)i * M_ + t] * k[(size_t)i * M_ + t];
  __syncthreads();
#pragma unroll
  for (int s = 128; s > 0; s >>= 1) {
    if (t < s) sd[t] += sd[t + s];
    __syncthreads();
  }
  const float pos_z = sd[0] * TEMP_INV;
  __syncthreads();

  // global max over partial maxima and pos_z
  const float m_t = pm[(size_t)i * NT_ + t];
  sd[t] = m_t;
  __syncthreads();
#pragma unroll
  for (int s = 128; s > 0; s >>= 1) {
    if (t < s) sd[t] = fmaxf(sd[t], sd[t + s]);
    __syncthreads();
  }
  const float Mx = fmaxf(sd[0], pos_z);
  __syncthreads();

  // rescaled sum
  sd[t] = ps[(size_t)i * NT_ + t] * __expf(m_t - Mx);
  __syncthreads();
#pragma unroll
  for (int s = 128; s > 0; s >>= 1) {
    if (t < s) sd[t] += sd[t + s];
    __syncthreads();
  }
  if (t == 0) {
    const float S = sd[0] + __expf(pos_z - Mx);
    loss[i] = __logf(S) + Mx - pos_z;
  }
}

// ---------------------------------------------------------------------------
// Mean over the 1024 per-row losses -> d_out[0]. Single block, fixed order.
// ---------------------------------------------------------------------------
__global__ __launch_bounds__(256)
void loss_mean(const float* __restrict__ loss, float* __restrict__ out) {
  __shared__ float sd[256];
  const int t = threadIdx.x;
  float s = 0.f;
#pragma unroll
  for (int j = 0; j < 4; ++j) s += loss[t * 4 + j];
  sd[t] = s;
  __syncthreads();
#pragma unroll
  for (int st = 128; st > 0; st >>= 1) {
    if (t < st) sd[t] += sd[t + st];
    __syncthreads();
  }
  if (t == 0) out[0] = sd[0] * (1.f / (float)B_);
}

// ---------------------------------------------------------------------------
// new_queue = queue (grid-stride copy into d_out+1; base is 4B-offset so we
// stay with scalar dwords), then scatter the normalized k rows at write_idx.
// ---------------------------------------------------------------------------
__global__ __launch_bounds__(256)
void queue_copy(const float* __restrict__ src, float* __restrict__ dst) {
  const size_t base = (size_t)blockIdx.x * 1024 + threadIdx.x;
#pragma unroll
  for (int j = 0; j < 4; ++j) {
    const size_t idx = base + (size_t)j * 256;
    dst[idx] = src[idx];
  }
}

__global__ __launch_bounds__(256)
void queue_scatter(const float* __restrict__ k, const int* __restrict__ widx,
                   float* __restrict__ dst) {
  const int i = blockIdx.x, t = threadIdx.x;
  const int row = widx[i];
  dst[(size_t)row * M_ + t] = k[(size_t)i * M_ + t];
}

// ---------------------------------------------------------------------------
extern "C" void kernel_launch(void* const* d_in, const int* in_sizes, int n_in,
                              void* d_out, int out_size, void* d_ws, size_t ws_size,
                              hipStream_t stream) {
  const float* y1    = (const float*)d_in[0];
  const float* y2    = (const float*)d_in[1];
  const float* Wq    = (const float*)d_in[2];
  const float* Wk    = (const float*)d_in[3];
  const float* Wp    = (const float*)d_in[4];
  const float* queue = (const float*)d_in[5];
  const int*   sidx  = (const int*)d_in[6];
  const int*   widx  = (const int*)d_in[7];
  float* out = (float*)d_out;

  // workspace layout (floats)
  float* ws   = (float*)d_ws;
  float* Z    = ws;                         // [1024,1024]   (reused q/k paths)
  float* qbuf = Z    + (size_t)B_ * H_;     // [1024,256]
  float* kbuf = qbuf + (size_t)B_ * M_;     // [1024,256]
  float* neg  = kbuf + (size_t)B_ * M_;     // [16384,256]
  float* pm   = neg  + (size_t)NNEG_ * M_;  // [1024,256] partial max
  float* ps   = pm   + (size_t)B_ * NT_;    // [1024,256] partial sumexp
  float* lbuf = ps   + (size_t)B_ * NT_;    // [1024]

  const dim3 blkW(32);

  // ---- query path: q = l2norm((y2 @ Wq) @ Wp)
  gemm_f32_wmma<<<dim3(H_ / 64, B_ / 16), blkW, 0, stream>>>(y2, Wq, Z, DIN_, H_);
  gemm_f32_wmma<<<dim3(M_ / 64, B_ / 16), blkW, 0, stream>>>(Z, Wp, qbuf, H_, M_);
  rownorm_inplace<<<B_, 256, 0, stream>>>(qbuf);

  // ---- key path: k = l2norm((y1 @ Wk) @ Wp)
  gemm_f32_wmma<<<dim3(H_ / 64, B_ / 16), blkW, 0, stream>>>(y1, Wk, Z, DIN_, H_);
  gemm_f32_wmma<<<dim3(M_ / 64, B_ / 16), blkW, 0, stream>>>(Z, Wp, kbuf, H_, M_);
  rownorm_inplace<<<B_, 256, 0, stream>>>(kbuf);

  // ---- negatives: gather + normalize
  gather_norm<<<NNEG_, 256, 0, stream>>>(queue, sidx, neg);

  // ---- fused logits + softmax partials (never materializes [1024,16384])
  logits_partial<<<dim3(NT_, B_ / 16), blkW, 0, stream>>>(qbuf, neg, pm, ps);

  // ---- per-row loss, then deterministic mean into d_out[0]
  row_loss<<<B_, 256, 0, stream>>>(qbuf, kbuf, pm, ps, lbuf);
  loss_mean<<<1, 256, 0, stream>>>(lbuf, out);

  // ---- new_queue = queue; new_queue[write_idx] = k   (d_out[1:])
  queue_copy<<<(Q_ * M_) / 1024, 256, 0, stream>>>(queue, out + 1);
  queue_scatter<<<B_, 256, 0, stream>>>(kbuf, widx, out + 1);
}